// SuppressBlock_55602646614361
// MI455X (gfx1250) — compile-verified
//
#include <hip/hip_runtime.h>

// ---------------- problem constants (match reference) ----------------
#define BATCH   16
#define NPTS    500000
#define KTOP    2000          // N_SCORE
#define NOUT    300           // N_NMS
#define NMS_THF 0.7f

// ---------------- algorithm constants ----------------
#define NBINS   4096          // linear value bins for top-k threshold
#define CAP     4096          // candidate cap per batch (expected ~2100)
#define MWORDS  64            // padded mask row: 64 dwords = 256 B
#define MROWS   2048          // padded row count (>= KTOP)
#define CHUNK_ROWS  256
#define CHUNK_BYTES (CHUNK_ROWS * MWORDS * 4)   // 65536 B per LDS chunk

// ---------------- workspace layout (bytes) ----------------
#define OFF_HIST   0                                    // 16*4096*4 = 262144
#define OFF_CNT    (OFF_HIST + BATCH * NBINS * 4)       // 262144
#define OFF_TH     (OFF_CNT + BATCH * 4)                // 262208
#define OFF_CAND   (OFF_TH + BATCH * 4)                 // 262272 (8B aligned)
#define OFF_SELBOX (OFF_CAND + BATCH * CAP * 8)         // 786560 (16B aligned)
#define OFF_MASK   1310720                              // 256B aligned
// total = OFF_MASK + 16*2048*64*4 = 9,699,328 bytes

// ---------------- CDNA5 helpers ----------------
// Low 32 bits of a generic pointer to LDS == LDS byte address (flat-LDS mapping).
__device__ __forceinline__ unsigned lds_lo32(const void* p) {
  return (unsigned)(unsigned long long)p;
}
// gfx1250 async global->LDS 128b copy (per-lane), tracked by ASYNCcnt.
__device__ __forceinline__ void async_b128(unsigned lds_off, const void* gptr) {
  unsigned long long ga = (unsigned long long)gptr;
  asm volatile("global_load_async_to_lds_b128 %0, %1, off"
               :: "v"(lds_off), "v"(ga) : "memory");
}
__device__ __forceinline__ void wait_async0() {
  asm volatile("s_wait_asynccnt 0x0" ::: "memory");
}
__device__ __forceinline__ void wait_ds0() {
  asm volatile("s_wait_dscnt 0x0" ::: "memory");
}

__device__ __forceinline__ int score_bin(float s) {
  int b = (int)(s * (float)NBINS);
  b = b < 0 ? 0 : b;
  b = b > NBINS - 1 ? NBINS - 1 : b;
  return b;
}

// ---------------- 1) per-batch score histogram ----------------
__global__ void sb_hist_kernel(const float* __restrict__ prob,
                               unsigned* __restrict__ hist) {
  __shared__ unsigned sh[NBINS];
  const int b = blockIdx.y;
  for (int t = threadIdx.x; t < NBINS; t += blockDim.x) sh[t] = 0u;
  __syncthreads();
  const float* p = prob + (size_t)b * NPTS;
  const int stride = gridDim.x * blockDim.x;
  for (int i = blockIdx.x * blockDim.x + threadIdx.x; i < NPTS; i += stride) {
    if (i + 8 * stride < NPTS) __builtin_prefetch(&p[i + 8 * stride], 0, 0);
    atomicAdd(&sh[score_bin(p[i])], 1u);
  }
  __syncthreads();
  unsigned* gh = hist + (size_t)b * NBINS;
  for (int t = threadIdx.x; t < NBINS; t += blockDim.x)
    if (sh[t]) atomicAdd(&gh[t], sh[t]);
}

// ---------------- 2) per-batch threshold bin ----------------
__global__ void sb_thresh_kernel(const unsigned* __restrict__ hist,
                                 int* __restrict__ thresh) {
  __shared__ unsigned sh[NBINS];
  const int b = blockIdx.x;
  const unsigned* gh = hist + (size_t)b * NBINS;
  for (int t = threadIdx.x; t < NBINS; t += blockDim.x) sh[t] = gh[t];
  __syncthreads();
  if (threadIdx.x == 0) {
    unsigned cum = 0;
    int T = 0;
    for (int t = NBINS - 1; t >= 0; --t) {
      cum += sh[t];
      if (cum >= KTOP) { T = t; break; }
    }
    thresh[b] = T;
  }
}

// ---------------- 3) collect candidates >= threshold bin ----------------
__global__ void sb_collect_kernel(const float* __restrict__ prob,
                                  const int* __restrict__ thresh,
                                  unsigned* __restrict__ cnt,
                                  unsigned long long* __restrict__ cand) {
  const int b = blockIdx.y;
  const int T = thresh[b];
  const float* p = prob + (size_t)b * NPTS;
  unsigned long long* cb = cand + (size_t)b * CAP;
  const int stride = gridDim.x * blockDim.x;
  for (int i = blockIdx.x * blockDim.x + threadIdx.x; i < NPTS; i += stride) {
    if (i + 8 * stride < NPTS) __builtin_prefetch(&p[i + 8 * stride], 0, 0);
    float s = p[i];
    if (score_bin(s) >= T) {
      unsigned pos = atomicAdd(&cnt[b], 1u);
      if (pos < CAP) {
        // key: score bits (positive floats are order-preserving) desc,
        // tie-break lower index first -> (0xFFFFFFFF - i) in low word.
        unsigned long long key =
            ((unsigned long long)__float_as_uint(s) << 32) |
            (unsigned long long)(0xFFFFFFFFu - (unsigned)i);
        cb[pos] = key;
      }
    }
  }
}

// ---------------- 4) bitonic sort (desc) + gather top-K boxes ----------------
__global__ void sb_sort_gather_kernel(const unsigned long long* __restrict__ cand,
                                      const unsigned* __restrict__ cnt,
                                      const float* __restrict__ roi,
                                      float4* __restrict__ selbox) {
  __shared__ unsigned long long sk[CAP];   // 32 KB
  const int b = blockIdx.x;
  unsigned n = cnt[b];
  if (n > CAP) n = CAP;
  const unsigned long long* cb = cand + (size_t)b * CAP;
  for (int i = threadIdx.x; i < CAP; i += blockDim.x)
    sk[i] = (i < (int)n) ? cb[i] : 0ULL;
  __syncthreads();
  for (int k = 2; k <= CAP; k <<= 1) {
    for (int j = k >> 1; j > 0; j >>= 1) {
      for (int i = threadIdx.x; i < CAP; i += blockDim.x) {
        int ixj = i ^ j;
        if (ixj > i) {
          unsigned long long a = sk[i], c = sk[ixj];
          bool up = ((i & k) == 0);         // descending overall
          if (up ? (a < c) : (a > c)) { sk[i] = c; sk[ixj] = a; }
        }
      }
      __syncthreads();
    }
  }
  const float4* rb = (const float4*)roi + (size_t)b * NPTS;
  float4* ob = selbox + (size_t)b * KTOP;
  for (int i = threadIdx.x; i < KTOP; i += blockDim.x) {
    unsigned long long key = sk[i];
    unsigned idx = 0xFFFFFFFFu - (unsigned)(key & 0xFFFFFFFFu);
    float4 v = make_float4(0.f, 0.f, 0.f, 0.f);
    if (key != 0ULL && idx < NPTS) v = rb[idx];
    ob[i] = v;
  }
}

// ---------------- 5) parallel IoU suppression bitmask ----------------
__global__ void sb_mask_kernel(const float4* __restrict__ selbox,
                               unsigned* __restrict__ mask) {
  __shared__ float4 sbox[KTOP];            // 32000 B, staged via async DMA
  const int b = blockIdx.y;
  const float4* bb = selbox + (size_t)b * KTOP;
  for (int j = threadIdx.x; j < KTOP; j += blockDim.x)
    async_b128(lds_lo32(&sbox[j]), &bb[j]);
  wait_async0();
  __syncthreads();

  const int i = blockIdx.x * blockDim.x + threadIdx.x;   // one row per thread
  if (i < KTOP) {
    float4 a = sbox[i];
    float aA = (a.z - a.x) * (a.w - a.y);
    unsigned* row = mask + ((size_t)b * MROWS + (size_t)i) * MWORDS;
    for (int w = 0; w < MWORDS; ++w) {
      unsigned bits = 0u;
      int j0 = w << 5;
      if (j0 + 31 > i && j0 < KTOP) {
        int jlo = (j0 > i + 1) ? j0 : i + 1;
        int jhi = (j0 + 32 < KTOP) ? j0 + 32 : KTOP;
        for (int j = jlo; j < jhi; ++j) {
          float4 c = sbox[j];
          float iy1 = fmaxf(a.x, c.x), ix1 = fmaxf(a.y, c.y);
          float iy2 = fminf(a.z, c.z), ix2 = fminf(a.w, c.w);
          float inter = fmaxf(iy2 - iy1, 0.f) * fmaxf(ix2 - ix1, 0.f);
          float cA = (c.z - c.x) * (c.w - c.y);
          float iou = inter / (aA + cA - inter);
          if (iou > NMS_THF) bits |= (1u << (j - j0));
        }
      }
      row[w] = bits;
    }
  }
}

// ---------------- 6) single-wave serial reduction, async double-buffered ----
__global__ void sb_nms_serial_kernel(const unsigned* __restrict__ mask,
                                     const float4* __restrict__ selbox,
                                     float* __restrict__ out) {
  extern __shared__ unsigned sbuf[];       // 2 * CHUNK_BYTES (128 KB)
  __shared__ unsigned sremv[64];
  __shared__ unsigned short skeep[KTOP];
  const int b = blockIdx.x;
  const int lane = threadIdx.x;            // 32 threads = one wave32
  volatile unsigned* vremv = sremv;
  vremv[lane] = 0u;
  vremv[lane + 32] = 0u;

  const char* mbase = (const char*)(mask + (size_t)b * MROWS * MWORDS);
  const unsigned lds0 = lds_lo32(&sbuf[0]);

  auto issue_chunk = [&](int c) {
    unsigned ldsbase = lds0 + (unsigned)((c & 1) * CHUNK_BYTES);
    const char* g = mbase + (size_t)c * CHUNK_BYTES;
    for (int o = lane * 16; o < CHUNK_BYTES; o += 32 * 16)
      async_b128(ldsbase + (unsigned)o, g + o);
  };

  const int nchunks = (KTOP + CHUNK_ROWS - 1) / CHUNK_ROWS;  // 8
  issue_chunk(0);
  wait_async0();

  int nk = 0;                               // identical in every lane
  for (int c = 0; c < nchunks; ++c) {
    if (c + 1 < nchunks) issue_chunk(c + 1);
    const unsigned base = (unsigned)((c & 1) * (CHUNK_BYTES / 4));
    const int i0 = c * CHUNK_ROWS;
    const int i1 = (i0 + CHUNK_ROWS < KTOP) ? i0 + CHUNK_ROWS : KTOP;
    for (int i = i0; i < i1; ++i) {
      unsigned rv = vremv[i >> 5];          // uniform LDS broadcast
      if (!((rv >> (i & 31)) & 1u)) {
        if (lane == 0 && nk < KTOP) skeep[nk] = (unsigned short)i;
        nk++;
        unsigned roff = base + (unsigned)(i - i0) * MWORDS;
        vremv[lane] |= sbuf[roff + lane];
        if (lane < 31) vremv[lane + 32] |= sbuf[roff + 32 + lane];
        wait_ds0();                         // wave-lockstep RAW on sremv
      }
    }
    wait_async0();                          // next chunk resident
  }
  wait_ds0();

  const float4* bb = selbox + (size_t)b * KTOP;
  float4* ob = (float4*)out + (size_t)b * NOUT;
  for (int r = lane; r < NOUT; r += 32) {
    float4 v = make_float4(0.f, 0.f, 0.f, 0.f);
    if (r < nk) v = bb[skeep[r]];
    ob[r] = v;                              // zero-padded, score order
  }
}

// ---------------- launch ----------------
extern "C" void kernel_launch(void* const* d_in, const int* in_sizes, int n_in,
                              void* d_out, int out_size, void* d_ws, size_t ws_size,
                              hipStream_t stream) {
  (void)in_sizes; (void)n_in; (void)out_size; (void)ws_size;
  const float* prob = (const float*)d_in[0];   // (16, 500000, 1) f32
  const float* roi  = (const float*)d_in[1];   // (16, 500000, 4) f32
  float* out = (float*)d_out;                  // (16, 300, 4) f32
  char* ws = (char*)d_ws;

  unsigned* hist            = (unsigned*)(ws + OFF_HIST);
  unsigned* cnt             = (unsigned*)(ws + OFF_CNT);
  int* thresh               = (int*)(ws + OFF_TH);
  unsigned long long* cand  = (unsigned long long*)(ws + OFF_CAND);
  float4* selbox            = (float4*)(ws + OFF_SELBOX);
  unsigned* mask            = (unsigned*)(ws + OFF_MASK);

  // zero hist + counters + thresholds every call (graph-capturable)
  hipMemsetAsync(ws, 0, OFF_CAND, stream);

  sb_hist_kernel<<<dim3(32, BATCH), 256, 0, stream>>>(prob, hist);
  sb_thresh_kernel<<<BATCH, 256, 0, stream>>>(hist, thresh);
  sb_collect_kernel<<<dim3(64, BATCH), 256, 0, stream>>>(prob, thresh, cnt, cand);
  sb_sort_gather_kernel<<<BATCH, 512, 0, stream>>>(cand, cnt, roi, selbox);
  sb_mask_kernel<<<dim3(MROWS / 256, BATCH), 256, 0, stream>>>(selbox, mask);
  sb_nms_serial_kernel<<<BATCH, 32, 2 * CHUNK_BYTES, stream>>>(mask, selbox, out);
}